// Attention_65798898975524
// MI455X (gfx1250) — compile-verified
//
#include <hip/hip_runtime.h>
#include <hip/hip_bf16.h>

// Linear attention, B=4 H=8 S=8192 D=64, fp32.
// Memory-bound (AI ~16 FLOP/B) -> full-precision V_WMMA_F32_16X16X4_F32,
// Tensor Data Mover (TDM) for the pass-through tiles (V chunk, KV matrix).

typedef __attribute__((ext_vector_type(2))) float v2f;
typedef __attribute__((ext_vector_type(8))) float v8f;
typedef __attribute__((ext_vector_type(4))) unsigned int u32x4;
typedef __attribute__((ext_vector_type(8))) int i32x8;
typedef __attribute__((ext_vector_type(4))) int i32x4;

#define S_LEN 8192
#define D_DIM 64
#define BH_N 32           // B*H
#define CHUNK 128         // rows staged in LDS per inner iteration
#define NCHUNK 8          // phase-1 blocks per (b,h)
#define S_PER_BLK (S_LEN / NCHUNK)   // 1024
#define LDS_STRIDE 65     // pad to spread LDS banks (matches TDM pad below)
#define PER_BH 4160       // 64*64 kv + 64 ksum floats

__device__ __forceinline__ float phi(float x) {
    // elu(x)+1 == (x>0 ? x+1 : exp(x))
    return x > 0.0f ? x + 1.0f : __expf(x);
}

// TDM 2D tile load: rows x 64 f32, row stride 64 elements in memory,
// LDS padded by 1 DWORD every 64 DWORDs (-> stride 65 floats).
// D# layout per CDNA5 ISA 8.3/8.4 (group0: count/lds/global/type,
// group1: data_size=4B, pad_enable, pad_interval=5 (64dw), pad_amount=0 (1dw),
// tensor_dim0=64, tensor_dim1=rows, tile_dim0=64, tile_dim1=rows, stride0=64).
// This toolchain exposes the 6-arg builtin:
//   (u32x4 g0, i32x8 g1, i32x4 g2, i32x4 g3, i32x8 g4, i32 cpol)
__device__ __forceinline__ void tdm_load_rows64(unsigned lds_off,
                                                const float* gptr,
                                                unsigned rows) {
  unsigned long long ga = (unsigned long long)(const void*)gptr;
  u32x4 g0;
  g0.x = 1u;                                        // count=1 (valid user desc)
  g0.y = lds_off;                                   // lds_addr (bytes)
  g0.z = (unsigned)(ga & 0xFFFFFFFFu);              // global_addr[31:0]
  g0.w = (unsigned)((ga >> 32) & 0x1FFFFFFu) | (2u << 30);  // [56:32] | type=2
  i32x8 g1 = {
      (int)((2u << 16) | (1u << 20) | (5u << 22)),  // 4B elems, pad 1dw/64dw
      (int)(64u << 16),                             // tensor_dim0 = 64
      (int)(rows << 16),                            // tensor_dim1 = rows
      (int)(64u << 16),                             // tile_dim0 = 64
      (int)rows,                                    // tile_dim1 = rows
      64,                                           // tensor_dim0_stride = 64
      0, 0};
  i32x4 z4 = {0, 0, 0, 0};
  i32x8 z8 = {0, 0, 0, 0, 0, 0, 0, 0};
  __builtin_amdgcn_tensor_load_to_lds(g0, g1, z4, z4, z8, 0);
}

// ---------------------------------------------------------------------------
// Phase 1: per (bh, chunk-block) partial KV = (phi(K)*m)^T * V and ksum
// (mask applied to K only: m^2 == m, and the 1/S * S factors cancel)
// ---------------------------------------------------------------------------
__global__ __launch_bounds__(256) void lin_attn_phase1(
    const float* __restrict__ kmat, const float* __restrict__ vmat,
    const unsigned char* __restrict__ kv_mask, float* __restrict__ part) {
  __shared__ float lk[CHUNK * LDS_STRIDE];
  __shared__ float lv[CHUNK * LDS_STRIDE];
  __shared__ float lsum[4 * D_DIM];

  const int bh  = blockIdx.y;
  const int b   = bh >> 3;                 // H == 8
  const int tid = threadIdx.x;             // 256 threads = 8 waves (wave32)
  const int wave = tid >> 5;
  const int lane = tid & 31;
  const int lhi  = lane >> 4;              // half-wave select
  const int llo  = lane & 15;

  // Each of 8 waves owns two 16x16 output tiles of the 64x64 KV matrix.
  const int t0 = wave, t1 = wave + 8;
  const int tm0 = t0 >> 2, tn0 = t0 & 3;
  const int tm1 = t1 >> 2, tn1 = t1 & 3;
  v8f acc0 = {}; v8f acc1 = {};
  float ksum_acc = 0.0f;
  const int sum_d     = tid & 63;
  const int sum_slice = tid >> 6;          // 4 slices of 32 rows

  const unsigned lv_off = (unsigned)(uintptr_t)(void*)lv;
  const int sBlk = blockIdx.x * S_PER_BLK;

  for (int cc = 0; cc < S_PER_BLK; cc += CHUNK) {
    const int s0 = sBlk + cc;
    if (cc) __syncthreads();               // protect LDS from prior readers

    // Wave-level TDM: DMA this wave's 16-row slice of V straight into LDS.
    tdm_load_rows64(lv_off + (unsigned)(wave * 16 * LDS_STRIDE * 4),
                    vmat + ((size_t)bh * S_LEN + s0 + wave * 16) * D_DIM, 16u);

    // Meanwhile, VALU stages phi(k)*mask (overlaps with the DMA).
    for (int i = tid; i < CHUNK * D_DIM; i += 256) {
      int r = i >> 6, c = i & 63;
      int s = s0 + r;
      float m  = kv_mask[b * S_LEN + s] ? 1.0f : 0.0f;
      float kx = kmat[((size_t)bh * S_LEN + s) * D_DIM + c];
      lk[r * LDS_STRIDE + c] = phi(kx) * m;
    }

    __builtin_amdgcn_s_wait_tensorcnt(0);
    __syncthreads();

    // Per-thread partial column sums of Ke.
    {
      float a = 0.0f;
      const int r0 = sum_slice * 32;
      #pragma unroll 8
      for (int r = r0; r < r0 + 32; ++r) a += lk[r * LDS_STRIDE + sum_d];
      ksum_acc += a;
    }

    // KV += Ke^T * V over this chunk. A = Ke^T tile (16x4), B = V tile (4x16).
    // A/C/D layouts per ISA 7.12.2; B assumed mirror of A (K = 2*lhi + vgpr).
    for (int kk = 0; kk < CHUNK; kk += 4) {
      const int rs = kk + 2 * lhi;
      v2f a0, a1, b0, b1;
      a0.x = lk[(rs    ) * LDS_STRIDE + tm0 * 16 + llo];
      a0.y = lk[(rs + 1) * LDS_STRIDE + tm0 * 16 + llo];
      a1.x = lk[(rs    ) * LDS_STRIDE + tm1 * 16 + llo];
      a1.y = lk[(rs + 1) * LDS_STRIDE + tm1 * 16 + llo];
      b0.x = lv[(rs    ) * LDS_STRIDE + tn0 * 16 + llo];
      b0.y = lv[(rs + 1) * LDS_STRIDE + tn0 * 16 + llo];
      b1.x = lv[(rs    ) * LDS_STRIDE + tn1 * 16 + llo];
      b1.y = lv[(rs + 1) * LDS_STRIDE + tn1 * 16 + llo];
      acc0 = __builtin_amdgcn_wmma_f32_16x16x4_f32(false, a0, false, b0,
                                                   (short)0, acc0, false, false);
      acc1 = __builtin_amdgcn_wmma_f32_16x16x4_f32(false, a1, false, b1,
                                                   (short)0, acc1, false, false);
    }
  }
  __syncthreads();

  // Deterministic (atomic-free) partial writeback.
  float* pbase = part + ((size_t)bh * NCHUNK + blockIdx.x) * PER_BH;
  #pragma unroll
  for (int i = 0; i < 8; ++i) {
    pbase[(tm0 * 16 + i + 8 * lhi) * D_DIM + tn0 * 16 + llo] = acc0[i];
    pbase[(tm1 * 16 + i + 8 * lhi) * D_DIM + tn1 * 16 + llo] = acc1[i];
  }

  lsum[sum_slice * D_DIM + sum_d] = ksum_acc;
  __syncthreads();
  if (tid < D_DIM) {
    float s = lsum[tid] + lsum[D_DIM + tid] + lsum[2 * D_DIM + tid] +
              lsum[3 * D_DIM + tid];
    pbase[4096 + tid] = s;
  }
}

// ---------------------------------------------------------------------------
// Reduce: sum NCHUNK partials per (bh) -> reduced kv/ksum
// ---------------------------------------------------------------------------
__global__ __launch_bounds__(256) void lin_attn_reduce(
    const float* __restrict__ part, float* __restrict__ red) {
  int i = blockIdx.x * blockDim.x + threadIdx.x;
  if (i >= BH_N * PER_BH) return;
  int bh = i / PER_BH, j = i - bh * PER_BH;
  float s = 0.0f;
  #pragma unroll
  for (int c = 0; c < NCHUNK; ++c)
    s += part[((size_t)bh * NCHUNK + c) * PER_BH + j];
  red[i] = s;
}

// ---------------------------------------------------------------------------
// Phase 2: out = (Qe @ KV) * div,  div = 1/(Qe . ksum + 1e-6)
// (S scale folded/cancelled against the 1/S omitted from KV)
// ---------------------------------------------------------------------------
__global__ __launch_bounds__(256) void lin_attn_phase2(
    const float* __restrict__ qmat, const unsigned char* __restrict__ q_mask,
    const float* __restrict__ red, float* __restrict__ out) {
  __shared__ float lq[CHUNK * LDS_STRIDE];
  __shared__ float lkv[D_DIM * LDS_STRIDE];
  __shared__ float lks[D_DIM];
  __shared__ float ldiv[CHUNK];

  const int bh  = blockIdx.y;
  const int b   = bh >> 3;
  const int q0  = blockIdx.x * CHUNK;
  const int tid = threadIdx.x;
  const int wave = tid >> 5, lane = tid & 31;
  const int lhi = lane >> 4, llo = lane & 15;

  // Wave-level TDM: DMA this wave's 8-row slice of KV into padded LDS.
  const float* rb = red + (size_t)bh * PER_BH;
  const unsigned lkv_off = (unsigned)(uintptr_t)(void*)lkv;
  tdm_load_rows64(lkv_off + (unsigned)(wave * 8 * LDS_STRIDE * 4),
                  rb + wave * 8 * D_DIM, 8u);

  if (tid < D_DIM) lks[tid] = rb[4096 + tid];

  // Stage phi(q)*mask (overlaps with the DMA).
  for (int i = tid; i < CHUNK * D_DIM; i += 256) {
    int r = i >> 6, c = i & 63;
    int s = q0 + r;
    float m = q_mask[b * S_LEN + s] ? 1.0f : 0.0f;
    float x = qmat[((size_t)bh * S_LEN + s) * D_DIM + c];
    lq[r * LDS_STRIDE + c] = phi(x) * m;
  }
  __builtin_amdgcn_s_wait_tensorcnt(0);
  __syncthreads();

  // Normalizer per query row.
  if (tid < CHUNK) {
    float dot = 0.0f;
    #pragma unroll 16
    for (int d = 0; d < D_DIM; ++d) dot += lq[tid * LDS_STRIDE + d] * lks[d];
    ldiv[tid] = 1.0f / (dot + 1e-6f);
  }
  __syncthreads();

  // Wave w owns the 16-row band w; 4 column tiles across D=64.
  v8f acc[4] = {};
  for (int kk = 0; kk < D_DIM; kk += 4) {
    const int kb = kk + 2 * lhi;
    v2f a;
    a.x = lq[(wave * 16 + llo) * LDS_STRIDE + kb];
    a.y = lq[(wave * 16 + llo) * LDS_STRIDE + kb + 1];
    #pragma unroll
    for (int tn = 0; tn < 4; ++tn) {
      v2f bf;
      bf.x = lkv[(kb    ) * LDS_STRIDE + tn * 16 + llo];
      bf.y = lkv[(kb + 1) * LDS_STRIDE + tn * 16 + llo];
      acc[tn] = __builtin_amdgcn_wmma_f32_16x16x4_f32(false, a, false, bf,
                                                      (short)0, acc[tn], false, false);
    }
  }

  #pragma unroll
  for (int tn = 0; tn < 4; ++tn) {
    #pragma unroll
    for (int i = 0; i < 8; ++i) {
      int row = wave * 16 + i + 8 * lhi;
      int col = tn * 16 + llo;
      out[((size_t)bh * S_LEN + q0 + row) * D_DIM + col] = acc[tn][i] * ldiv[row];
    }
  }
}

// ---------------------------------------------------------------------------
extern "C" void kernel_launch(void* const* d_in, const int* in_sizes, int n_in,
                              void* d_out, int out_size, void* d_ws, size_t ws_size,
                              hipStream_t stream) {
  (void)in_sizes; (void)n_in; (void)out_size; (void)ws_size;
  const float* q = (const float*)d_in[0];
  const float* k = (const float*)d_in[1];
  const float* v = (const float*)d_in[2];
  const unsigned char* q_mask  = (const unsigned char*)d_in[3];
  const unsigned char* kv_mask = (const unsigned char*)d_in[4];
  float* out = (float*)d_out;

  // Workspace: [partials: BH*NCHUNK*PER_BH] [reduced: BH*PER_BH]
  float* part = (float*)d_ws;
  float* red  = part + (size_t)BH_N * NCHUNK * PER_BH;

  dim3 g1(NCHUNK, BH_N);                       // 8 x 32 blocks
  lin_attn_phase1<<<g1, 256, 0, stream>>>(k, v, kv_mask, part);

  int nred = BH_N * PER_BH;
  lin_attn_reduce<<<(nred + 255) / 256, 256, 0, stream>>>(part, red);

  dim3 g2(S_LEN / CHUNK, BH_N);                // 64 x 32 blocks
  lin_attn_phase2<<<g2, 256, 0, stream>>>(q, q_mask, red, out);
}